// SelfAttention_63376537420540
// MI455X (gfx1250) — compile-verified
//
#include <hip/hip_runtime.h>

typedef __attribute__((ext_vector_type(2))) float v2f;
typedef __attribute__((ext_vector_type(8))) float v8f;

#define NROWS 8192
#define DIM   64

// ---------------------------------------------------------------------------
// Zero the workspace region used for G (64x64) and c (64).
// ---------------------------------------------------------------------------
__global__ void zero_ws_kernel(float* __restrict__ ws, int n) {
  int i = blockIdx.x * blockDim.x + threadIdx.x;
  if (i < n) ws[i] = 0.0f;
}

// ---------------------------------------------------------------------------
// Phase 1: G = S^T S  (64x64, fp32)  and  c = column-sums of S (64 floats).
// One wave (32 lanes) per block; each block reduces `rows_per_block` rows of S
// into a private 4x4 grid of 16x16 f32 WMMA accumulators, then atomically
// adds partials into G / c.
//
// f32 WMMA 16x16x4 operand layout (wave32):
//   A (16x4): lane l in [0,16) => M=l ; v.x = K=2h, v.y = K=2h+1  (h=lane>=16)
//   B (4x16): N=l ; v.x = row K=2h, v.y = row K=2h+1
//   For G = S^T S both operand tiles have the SAME lane layout, so one load
//   of S[k..k+3][t*16+l] serves as both A-tile t and B-tile t.
// ---------------------------------------------------------------------------
__global__ __launch_bounds__(32) void gram_kernel(
    const float* __restrict__ S, float* __restrict__ G, float* __restrict__ c,
    int rows_per_block) {
  const int lane = threadIdx.x & 31;
  const int l = lane & 15;
  const int h = lane >> 4;
  const int row0 = blockIdx.x * rows_per_block;

  v8f acc[4][4] = {};
  float csum[4] = {0.f, 0.f, 0.f, 0.f};

  for (int k0 = 0; k0 < rows_per_block; k0 += 4) {
    const float* Sk = S + (size_t)(row0 + k0 + 2 * h) * DIM;
    v2f a[4];
#pragma unroll
    for (int t = 0; t < 4; ++t) {
      a[t].x = Sk[t * 16 + l];          // row k0+2h
      a[t].y = Sk[DIM + t * 16 + l];    // row k0+2h+1
      csum[t] += a[t].x + a[t].y;
    }
#pragma unroll
    for (int m = 0; m < 4; ++m)
#pragma unroll
      for (int n = 0; n < 4; ++n)
        acc[m][n] = __builtin_amdgcn_wmma_f32_16x16x4_f32(
            false, a[m], false, a[n], (short)0, acc[m][n], false, false);
  }

  // Scatter partial G (each lane owns 8 rows per tile: row = m*16 + r + 8h).
#pragma unroll
  for (int m = 0; m < 4; ++m)
#pragma unroll
    for (int n = 0; n < 4; ++n)
#pragma unroll
      for (int r = 0; r < 8; ++r) {
        int row = m * 16 + r + 8 * h;
        int col = n * 16 + l;
        atomicAdd(&G[row * DIM + col], acc[m][n][r]);
      }
  // Column sums: lane pair (l, l+16) both contribute to column t*16+l.
#pragma unroll
  for (int t = 0; t < 4; ++t) atomicAdd(&c[t * 16 + l], csum[t]);
}

// ---------------------------------------------------------------------------
// Phase 2: out = (S @ G) / (S @ c), one 16-row tile per block (one wave).
//   numerator tile: A(16x64) @ G(64x64) -> 4 WMMA accumulators of 16x16
//   denominator:    per-row dot(s_i, c), split across half-waves + shfl_xor
// ---------------------------------------------------------------------------
__global__ __launch_bounds__(32) void apply_kernel(
    const float* __restrict__ S, const float* __restrict__ G,
    const float* __restrict__ c, float* __restrict__ out) {
  const int lane = threadIdx.x & 31;
  const int l = lane & 15;
  const int h = lane >> 4;
  const int row0 = blockIdx.x * 16;
  const float* Arow = S + (size_t)(row0 + l) * DIM;

  v8f acc[4] = {};
#pragma unroll
  for (int k0 = 0; k0 < DIM; k0 += 4) {
    v2f a;
    a.x = Arow[k0 + 2 * h];        // A[M=l][K=2h]   (contiguous b64 load)
    a.y = Arow[k0 + 2 * h + 1];    // A[M=l][K=2h+1]
    const float* Gk = G + (size_t)(k0 + 2 * h) * DIM;
#pragma unroll
    for (int n = 0; n < 4; ++n) {
      v2f b;
      b.x = Gk[n * 16 + l];        // B[K=2h]  [N=l]
      b.y = Gk[DIM + n * 16 + l];  // B[K=2h+1][N=l]
      acc[n] = __builtin_amdgcn_wmma_f32_16x16x4_f32(
          false, a, false, b, (short)0, acc[n], false, false);
    }
  }

  // Denominator for row (row0 + l): each half-wave does half the dot product.
  float dpart = 0.f;
#pragma unroll
  for (int d = 0; d < 32; ++d) {
    int dd = 32 * h + d;
    dpart += Arow[dd] * c[dd];
  }
  float denom = dpart + __shfl_xor(dpart, 16, 32);  // lane l holds row l's denom

  // Store: D VGPR r holds row (r + 8h), col l of each 16-col tile.
#pragma unroll
  for (int r = 0; r < 8; ++r) {
    float dr = __shfl(denom, r + 8 * h, 32);
    float inv = 1.0f / dr;
    int row = row0 + r + 8 * h;
#pragma unroll
    for (int n = 0; n < 4; ++n)
      out[(size_t)row * DIM + n * 16 + l] = acc[n][r] * inv;
  }
}

// ---------------------------------------------------------------------------
extern "C" void kernel_launch(void* const* d_in, const int* in_sizes, int n_in,
                              void* d_out, int out_size, void* d_ws, size_t ws_size,
                              hipStream_t stream) {
  (void)in_sizes; (void)n_in; (void)out_size; (void)ws_size;
  const float* S = (const float*)d_in[0];
  float* out = (float*)d_out;
  float* G = (float*)d_ws;          // 64*64 floats
  float* c = G + DIM * DIM;         // 64 floats

  const int ws_elems = DIM * DIM + DIM;  // 4160
  zero_ws_kernel<<<(ws_elems + 255) / 256, 256, 0, stream>>>((float*)d_ws, ws_elems);

  const int blocks1 = 128;                       // 64 rows each
  gram_kernel<<<blocks1, 32, 0, stream>>>(S, G, c, NROWS / blocks1);

  apply_kernel<<<NROWS / 16, 32, 0, stream>>>(S, G, c, out);
}